// DCNv2_45973329936715
// MI455X (gfx1250) — compile-verified
//
#include <hip/hip_runtime.h>
#include <hip/hip_bf16.h>

typedef __attribute__((ext_vector_type(2))) float v2f;
typedef __attribute__((ext_vector_type(4))) float v4f;
typedef __attribute__((ext_vector_type(8))) float v8f;

#define B_   8
#define CIN  64
#define COUT 64
#define H_   128
#define W_   128
#define KK   9          // 3x3 taps
#define HW   (H_*W_)
#define OSTR 20         // padded LDS stride for output staging (16B aligned)

// ---------------------------------------------------------------------------
// Kernel 0a: pack conv_w [64,64,3,3] into WMMA B-matrix (4x16 f32) lane layout.
//   lane L: VGPR0 = W[o, c4+(L>>4), kpos], VGPR1 = W[o, c4+2+(L>>4), kpos]
//   o = nt*16 + (L&15), c4 = kc*4.  4 N-tiles (64 out-ch).
// ---------------------------------------------------------------------------
__global__ void dcn_pack_w(const float* __restrict__ Wc, float* __restrict__ Bp) {
  int t = blockIdx.x * blockDim.x + threadIdx.x;
  if (t >= 9 * 16 * 4 * 32) return;
  int lane = t & 31;
  int nt   = (t >> 5) & 3;
  int kc   = (t >> 7) & 15;
  int kpos = t >> 11;
  int o = nt * 16 + (lane & 15);
  int c = kc * 4 + (lane >> 4);
  Bp[2 * t + 0] = Wc[(o * CIN + c    ) * KK + kpos];
  Bp[2 * t + 1] = Wc[(o * CIN + c + 2) * KK + kpos];
}

// ---------------------------------------------------------------------------
// Kernel 0b: pack offset_w (18ch) + mask_w (9ch) into one 27->32 padded
// B-matrix stream, same lane layout, 2 N-tiles.
// ---------------------------------------------------------------------------
__global__ void dcn_pack_om(const float* __restrict__ ow, const float* __restrict__ mw,
                            float* __restrict__ Bp2) {
  int t = blockIdx.x * blockDim.x + threadIdx.x;
  if (t >= 9 * 16 * 2 * 32) return;
  int lane = t & 31;
  int nt   = (t >> 5) & 1;
  int kc   = (t >> 6) & 15;
  int kpos = t >> 10;
  int o = nt * 16 + (lane & 15);
  int c = kc * 4 + (lane >> 4);
  float w0 = 0.f, w1 = 0.f;
  if (o < 18) {
    w0 = ow[(o * CIN + c    ) * KK + kpos];
    w1 = ow[(o * CIN + c + 2) * KK + kpos];
  } else if (o < 27) {
    w0 = mw[((o - 18) * CIN + c    ) * KK + kpos];
    w1 = mw[((o - 18) * CIN + c + 2) * KK + kpos];
  }
  Bp2[2 * t + 0] = w0;
  Bp2[2 * t + 1] = w1;
}

// ---------------------------------------------------------------------------
// Kernel 1: offset(18ch) + mask(9ch) 3x3 conv as implicit-GEMM WMMA.
// One wave32 per 16-pixel strip: M=16 px, N=32 (27 used), K=576.
// A operand built directly in registers in WMMA-A lane layout:
//   lane L holds channels {4kc + 2*(L>>4), +1} for row m = L&15.
// OOB rows skip the tap (zero contribution).  No LDS at all.
// ---------------------------------------------------------------------------
__global__ __launch_bounds__(32) void dcn_offmask(
    const float* __restrict__ x,
    const float* __restrict__ Bp2,
    const float* __restrict__ ob, const float* __restrict__ mb,
    float* __restrict__ off, float* __restrict__ msk) {
  int lane = threadIdx.x;
  int blk  = blockIdx.x;
  int b  = blk >> 10;
  int h  = (blk >> 3) & 127;
  int w0 = (blk & 7) << 4;

  int pxl  = lane & 15;
  int half = lane >> 4;
  const float* xb = x + (size_t)b * CIN * HW;

  v8f acc[2] = {};
  for (int kpos = 0; kpos < 9; ++kpos) {
    int ky = kpos / 3, kx = kpos % 3;
    int y = h - 1 + ky;                        // wave-uniform
    if ((unsigned)y >= (unsigned)H_) continue; // zero tap: skip entirely
    int xx = w0 + pxl - 1 + kx;
    bool vx = (unsigned)xx < (unsigned)W_;
    int xcol = min(max(xx, 0), W_ - 1);
    size_t rowoff = (size_t)y * W_ + xcol;

    v2f a[16];
#pragma unroll
    for (int kc = 0; kc < 16; ++kc) {
      int c0 = kc * 4 + half * 2;
      const float* xc0 = xb + (size_t)c0 * HW + rowoff;
      a[kc].x = vx ? xc0[0]  : 0.f;
      a[kc].y = vx ? xc0[HW] : 0.f;
    }
    const float* bp = Bp2 + (size_t)kpos * 16 * 2 * 64;
#pragma unroll
    for (int kc = 0; kc < 16; ++kc) {
      const float* bpk = bp + kc * 2 * 64;
#pragma unroll
      for (int nt = 0; nt < 2; ++nt) {
        v2f bm = *(const v2f*)(bpk + nt * 64 + lane * 2);
        acc[nt] = __builtin_amdgcn_wmma_f32_16x16x4_f32(
            false, a[kc], false, bm, (short)0, acc[nt], false, false);
      }
    }
  }

  // epilogue: ch 0..17 -> offset (+ob), ch 18..26 -> sigmoid mask (+mb)
  int n = lane & 15;
  size_t pixb = (size_t)h * W_ + w0;
#pragma unroll
  for (int nt = 0; nt < 2; ++nt) {
    int ch = nt * 16 + n;
#pragma unroll
    for (int r = 0; r < 8; ++r) {
      int m = r + half * 8;
      float v = acc[nt][r];
      if (ch < 18) {
        off[(size_t)(b * 18 + ch) * HW + pixb + m] = v + ob[ch];
      } else if (ch < 27) {
        float s = v + mb[ch - 18];
        msk[(size_t)(b * 9 + ch - 18) * HW + pixb + m] = 1.f / (1.f + __expf(-s));
      }
    }
  }
}

// ---------------------------------------------------------------------------
// Kernel 2: deformable conv main loop.  One wave32 per 16-pixel strip.
// M=16 px, N=64 out-ch (4 tiles), K=576 via v_wmma_f32_16x16x4_f32.
// Bilinear samples gathered straight into the WMMA-A register layout
// (v2f a[16] per tap) -- no LDS round trip, taps pipeline through loadcnt.
// Epilogue transposes D through LDS -> 8 coalesced global_store_b128.
// ---------------------------------------------------------------------------
__global__ __launch_bounds__(32) void dcn_main(
    const float* __restrict__ x,  const float* __restrict__ off,
    const float* __restrict__ msk, const float* __restrict__ Bp,
    const float* __restrict__ bias, float* __restrict__ out) {
  __shared__ int   Py0[144], Px0[144], Py1[144], Px1[144];
  __shared__ float Pw0[144], Pw1[144], Pw2[144], Pw3[144];
  __shared__ float Dst[64 * OSTR];                  // output staging

  int lane = threadIdx.x;
  int blk  = blockIdx.x;
  int b  = blk >> 10;
  int h  = (blk >> 3) & 127;
  int w0 = (blk & 7) << 4;

  // -------- phase 1: per-(px,tap) bilinear params (shared by all 64 ch) ----
  for (int i = lane; i < 144; i += 32) {
    int k = i >> 4, px = i & 15;
    int w = w0 + px;
    size_t pix = (size_t)h * W_ + w;
    float oy = off[(size_t)(b * 18 + 2 * k    ) * HW + pix];
    float ox = off[(size_t)(b * 18 + 2 * k + 1) * HW + pix];
    float mk = msk[(size_t)(b * 9  + k        ) * HW + pix];
    float sy = (float)(h - 1 + k / 3) + oy;
    float sx = (float)(w - 1 + k % 3) + ox;
    float y0f = floorf(sy), x0f = floorf(sx);
    float wy = sy - y0f, wx = sx - x0f;
    int y0 = (int)y0f, x0 = (int)x0f;
    int y1 = y0 + 1,   x1 = x0 + 1;
    float vy0 = ((unsigned)y0 < (unsigned)H_) ? 1.f : 0.f;
    float vy1 = ((unsigned)y1 < (unsigned)H_) ? 1.f : 0.f;
    float vx0 = ((unsigned)x0 < (unsigned)W_) ? 1.f : 0.f;
    float vx1 = ((unsigned)x1 < (unsigned)W_) ? 1.f : 0.f;
    Pw0[i] = (1.f - wy) * (1.f - wx) * mk * vy0 * vx0;
    Pw1[i] = (1.f - wy) * wx         * mk * vy0 * vx1;
    Pw2[i] = wy         * (1.f - wx) * mk * vy1 * vx0;
    Pw3[i] = wy         * wx         * mk * vy1 * vx1;
    Py0[i] = min(max(y0, 0), H_ - 1);
    Py1[i] = min(max(y1, 0), H_ - 1);
    Px0[i] = min(max(x0, 0), W_ - 1);
    Px1[i] = min(max(x1, 0), W_ - 1);
  }
  // single-wave workgroup: per-wave s_wait_dscnt orders LDS use

  v8f acc[4] = {};
  const float* xb = x + (size_t)b * CIN * HW;
  int pxl  = lane & 15;
  int half = lane >> 4;

  for (int kpos = 0; kpos < 9; ++kpos) {
    // ---- gather 16px x 64ch bilinear samples directly into A registers ----
    int p = kpos * 16 + pxl;
    int a00 = Py0[p] * W_ + Px0[p];
    int a01 = Py0[p] * W_ + Px1[p];
    int a10 = Py1[p] * W_ + Px0[p];
    int a11 = Py1[p] * W_ + Px1[p];
    float w00 = Pw0[p], w01 = Pw1[p], w10 = Pw2[p], w11 = Pw3[p];

    v2f a[16];
#pragma unroll
    for (int kc = 0; kc < 16; ++kc) {
      int c0 = kc * 4 + half * 2;
      const float* xc0 = xb + (size_t)c0 * HW;
      const float* xc1 = xc0 + HW;
      a[kc].x = w00 * xc0[a00] + w01 * xc0[a01] + w10 * xc0[a10] + w11 * xc0[a11];
      a[kc].y = w00 * xc1[a00] + w01 * xc1[a01] + w10 * xc1[a10] + w11 * xc1[a11];
    }
    // ---- 16 K-chunks x 4 N-tiles of f32 WMMA ----
    const float* bp = Bp + (size_t)kpos * 16 * 4 * 64;
#pragma unroll
    for (int kc = 0; kc < 16; ++kc) {
      const float* bpk = bp + kc * 4 * 64;
#pragma unroll
      for (int nt = 0; nt < 4; ++nt) {
        v2f bm = *(const v2f*)(bpk + nt * 64 + lane * 2);
        acc[nt] = __builtin_amdgcn_wmma_f32_16x16x4_f32(
            false, a[kc], false, bm, (short)0, acc[nt], false, false);
      }
    }
  }

  // -------- epilogue: stage D in LDS [o][px], then coalesced b128 stores ---
  int n = lane & 15;
#pragma unroll
  for (int nt = 0; nt < 4; ++nt) {
    int o = nt * 16 + n;
#pragma unroll
    for (int r = 0; r < 8; ++r)
      Dst[o * OSTR + r + half * 8] = acc[nt][r];
  }
  size_t outb = (size_t)b * COUT * HW + (size_t)h * W_ + w0;
#pragma unroll
  for (int j = 0; j < 8; ++j) {
    int flat = j * 32 + lane;          // 256 quads = 64 rows x 4
    int o = flat >> 2, q = flat & 3;
    v4f v = *(const v4f*)&Dst[o * OSTR + q * 4];
    float bv = bias[o];
    v.x += bv; v.y += bv; v.z += bv; v.w += bv;
    *(v4f*)(out + outb + (size_t)o * HW + q * 4) = v;
  }
}

// ---------------------------------------------------------------------------
extern "C" void kernel_launch(void* const* d_in, const int* in_sizes, int n_in,
                              void* d_out, int out_size, void* d_ws, size_t ws_size,
                              hipStream_t stream) {
  const float* x  = (const float*)d_in[0];
  const float* ow = (const float*)d_in[1];
  const float* ob = (const float*)d_in[2];
  const float* mw = (const float*)d_in[3];
  const float* mb = (const float*)d_in[4];
  const float* cw = (const float*)d_in[5];
  const float* cb = (const float*)d_in[6];
  float* out = (float*)d_out;

  float* ws   = (float*)d_ws;
  float* Bp   = ws;                                  // 36864 f32
  float* Bp2  = Bp + 36864;                          // 18432 f32
  float* off  = Bp2 + 18432;                         // 8*18*128*128 f32
  float* msk  = off + (size_t)B_ * 18 * HW;          // 8*9*128*128 f32

  hipLaunchKernelGGL(dcn_pack_w,  dim3(72),   dim3(256), 0, stream, cw, Bp);
  hipLaunchKernelGGL(dcn_pack_om, dim3(36),   dim3(256), 0, stream, ow, mw, Bp2);
  hipLaunchKernelGGL(dcn_offmask, dim3(8192), dim3(32),  0, stream,
                     x, Bp2, ob, mb, off, msk);
  hipLaunchKernelGGL(dcn_main,    dim3(8192), dim3(32),  0, stream,
                     x, off, msk, Bp, cb, out);
}